// LiquidS4StateLoop_36232344109585
// MI455X (gfx1250) — compile-verified
//
#include <hip/hip_runtime.h>
#include <math.h>

// Problem constants (from reference): B=8, S=2048, D=1024, N=256
constexpr int kB = 8;
constexpr int kS = 2048;
constexpr int kD = 1024;
constexpr int kN = 256;
constexpr int kNCH = 16;               // scan chunks along S
constexpr int kCL = kS / kNCH;         // 128 steps per chunk

typedef __attribute__((ext_vector_type(2))) float v2f;
typedef __attribute__((ext_vector_type(8))) float v8f;

// ---------------------------------------------------------------------------
// CDNA5 async Global->LDS DMA (ASYNCcnt-tracked). The ROCm builtin takes
// (int4 addrspace(1)* src, int4 addrspace(3)* dst, imm offset, imm cpol).
// ---------------------------------------------------------------------------
#if defined(__has_builtin)
#if __has_builtin(__builtin_amdgcn_global_load_async_to_lds_b128)
#define HAVE_ASYNC_LDS_BUILTIN 1
#endif
#if __has_builtin(__builtin_amdgcn_s_wait_asynccnt)
#define HAVE_WAIT_ASYNC_BUILTIN 1
#endif
#endif

typedef int v4i_gcc __attribute__((vector_size(16)));
typedef __attribute__((address_space(1))) v4i_gcc* gas_v4i_ptr;
typedef __attribute__((address_space(3))) v4i_gcc* las_v4i_ptr;

__device__ __forceinline__ void async_copy_b128(const float* g, float* l) {
#ifdef HAVE_ASYNC_LDS_BUILTIN
    __builtin_amdgcn_global_load_async_to_lds_b128(
        (gas_v4i_ptr)(g), (las_v4i_ptr)(l), 0, 0);
#else
    asm volatile("global_load_async_to_lds_b128 %0, %1, off"
                 :: "v"((unsigned)(unsigned long long)l), "v"(g)
                 : "memory");
#endif
}

template <int N>
__device__ __forceinline__ void wait_asynccnt() {
#ifdef HAVE_WAIT_ASYNC_BUILTIN
    __builtin_amdgcn_s_wait_asynccnt((unsigned short)N);
#else
    asm volatile("s_wait_asynccnt %0" :: "n"(N) : "memory");
#endif
}

// -----------------------------------------------------------------------------
// WMMA f32 GEMM:  C[M,N] = A[M,K] * W[N,K]^T + bias[N]
//   - block tile 128(M) x 128(N), K-block 32, double-buffered LDS via async DMA
//   - 256 threads = 8 wave32s arranged 4(M) x 2(N); each wave: 32x64 sub-tile
//     = 2x4 grid of 16x16 v_wmma_f32_16x16x4_f32 accumulators (v8f each)
//   - LDS pitch 36 floats (144B: 16B-aligned rows, bank-conflict-free stride)
// Fragment layout per CDNA5 ISA (32-bit A 16x4): lanes 0-15 hold K=0,1 in the
// two VGPRs, lanes 16-31 hold K=2,3; B (4x16) mirrors this with N across lanes.
// -----------------------------------------------------------------------------
__global__ __launch_bounds__(256)
void gemm_f32_wmma(const float* __restrict__ A,
                   const float* __restrict__ W,
                   const float* __restrict__ bias,
                   float* __restrict__ C,
                   int M, int N, int K)
{
    constexpr int BM = 128, BN = 128, BK = 32, PITCH = BK + 4;
    __shared__ float As[2][BM * PITCH];
    __shared__ float Ws[2][BN * PITCH];

    const int tid  = threadIdx.x;
    const int lane = tid & 31;
    const int wave = tid >> 5;       // 0..7
    const int wm   = wave & 3;       // wave row: 32 M-rows each
    const int wn   = wave >> 2;      // wave col: 64 N-cols each
    const int l15  = lane & 15;
    const int lh   = lane >> 4;      // half-wave select (K pair / M+8)

    const int m0 = blockIdx.x * BM;
    const int n0 = blockIdx.y * BN;

    v8f acc[2][4];
    const v8f vzero = {};
#pragma unroll
    for (int i = 0; i < 2; ++i)
#pragma unroll
        for (int j = 0; j < 4; ++j) acc[i][j] = vzero;

    const int ldr = tid >> 3;        // 0..31 (row within 32-row slab)
    const int ldc = (tid & 7) * 4;   // float4 column

    // Stage one K-block into LDS buffer `bufi` with async DMA (8 copies/thread).
    auto stage = [&](int kb, int bufi) {
        const int k0 = kb * BK;
#pragma unroll
        for (int p = 0; p < 4; ++p) {
            const int r = ldr + p * 32;
            async_copy_b128(A + (size_t)(m0 + r) * K + k0 + ldc,
                            &As[bufi][r * PITCH + ldc]);
            async_copy_b128(W + (size_t)(n0 + r) * K + k0 + ldc,
                            &Ws[bufi][r * PITCH + ldc]);
        }
    };

    const int KB = K / BK;
    stage(0, 0);

    for (int kb = 0; kb < KB; ++kb) {
        const int cur = kb & 1;
        if (kb + 1 < KB) {
            stage(kb + 1, cur ^ 1);   // prefetch next block (overlaps compute)
            wait_asynccnt<8>();       // in-order completion => cur block done
        } else {
            wait_asynccnt<0>();
        }
        __syncthreads();              // all waves' DMA for `cur` visible

#pragma unroll
        for (int kk = 0; kk < BK; kk += 4) {
            const int kf = kk + lh * 2;          // K pair for this half-wave
            v2f afr[2], bfr[4];
#pragma unroll
            for (int tm = 0; tm < 2; ++tm)
                afr[tm] = *reinterpret_cast<const v2f*>(
                    &As[cur][(wm * 32 + tm * 16 + l15) * PITCH + kf]);
#pragma unroll
            for (int tn = 0; tn < 4; ++tn)
                bfr[tn] = *reinterpret_cast<const v2f*>(
                    &Ws[cur][(wn * 64 + tn * 16 + l15) * PITCH + kf]);
#pragma unroll
            for (int tm = 0; tm < 2; ++tm)
#pragma unroll
                for (int tn = 0; tn < 4; ++tn)
                    acc[tm][tn] = __builtin_amdgcn_wmma_f32_16x16x4_f32(
                        false, afr[tm], false, bfr[tn],
                        (short)0, acc[tm][tn], false, false);
        }
        __syncthreads();  // all waves done reading `cur` before it is re-staged
    }

    // Epilogue: C/D layout — VGPR j: lanes 0-15 -> M=j, lanes 16-31 -> M=j+8
#pragma unroll
    for (int tm = 0; tm < 2; ++tm) {
#pragma unroll
        for (int tn = 0; tn < 4; ++tn) {
            const int cg = n0 + wn * 64 + tn * 16 + l15;
            const float bv = bias[cg];
#pragma unroll
            for (int j = 0; j < 8; ++j) {
                const int rg = m0 + wm * 32 + tm * 16 + lh * 8 + j;
                C[(size_t)rg * N + cg] = acc[tm][tn][j] + bv;
            }
        }
    }
}

// -----------------------------------------------------------------------------
// Scan: r[s] = A*r[s-1] + Ball[s] (with r[0] = state0 + Ball[0]), decomposed as
// r[s] = z[s] + A^s*state0, z = zero-init scan, chunked into 16 chunks of 128.
// -----------------------------------------------------------------------------
__device__ __forceinline__ float sigmoidf_(float x) {
    return 1.0f / (1.0f + __expf(-x));
}

// Pass 1: in-place chunk-local zero-init scans; record per-chunk totals.
// grid = B*NCH blocks, block = N threads (coalesced along n).
__global__ void scan_local(const float* __restrict__ A_log,
                           float* __restrict__ buf,     // [B][S][N] Ball -> y
                           float* __restrict__ yend)    // [B][NCH][N]
{
    const int n  = threadIdx.x;
    const int bc = blockIdx.x;             // b*NCH + chunk
    const float a = sigmoidf_(A_log[n]);
    size_t base = ((size_t)bc) * kCL * kN + n;   // (b*S + c*CL)*N + n
    float y = 0.0f;
#pragma unroll 4
    for (int i = 0; i < kCL; ++i) {
        const float v = buf[base + (size_t)i * kN];
        y = fmaf(a, y, v);
        buf[base + (size_t)i * kN] = y;
    }
    yend[(size_t)bc * kN + n] = y;
}

// Pass 2: carry scan across chunk boundaries: e[c+1] = A^CL * e[c] + yend[c].
__global__ void scan_carry(const float* __restrict__ A_log,
                           const float* __restrict__ yend,  // [B][NCH][N]
                           float* __restrict__ ebuf)        // [B][NCH][N]
{
    const int idx = blockIdx.x * blockDim.x + threadIdx.x;  // b*N + n
    const int b = idx >> 8;
    const int n = idx & (kN - 1);
    const float a  = sigmoidf_(A_log[n]);
    const float aL = __powf(a, (float)kCL);
    float e = 0.0f;
    for (int c = 0; c < kNCH; ++c) {
        const size_t off = ((size_t)(b * kNCH + c)) * kN + n;
        ebuf[off] = e;
        e = fmaf(aL, e, yend[off]);
    }
}

// Pass 3: r[s] = y[s] + A^{s-cL} * (A*e[c] + A^{cL}*state0); also final_state.
__global__ void scan_apply(const float* __restrict__ A_log,
                           const float* __restrict__ state0,   // [B][N]
                           const float* __restrict__ ebuf,     // [B][NCH][N]
                           float* __restrict__ buf,            // y -> r in place
                           float* __restrict__ final_state)    // [B][N]
{
    const int n  = threadIdx.x;
    const int bc = blockIdx.x;
    const int b  = bc / kNCH;
    const int c  = bc % kNCH;
    const float a   = sigmoidf_(A_log[n]);
    const float acL = __powf(a, (float)(c * kCL));
    float w = a * ebuf[(size_t)bc * kN + n] + acL * state0[b * kN + n];
    size_t base = ((size_t)bc) * kCL * kN + n;
    float r = 0.0f;
#pragma unroll 4
    for (int i = 0; i < kCL; ++i) {
        r = buf[base + (size_t)i * kN] + w;
        buf[base + (size_t)i * kN] = r;
        w *= a;
    }
    if (c == kNCH - 1) final_state[b * kN + n] = r;
}

// -----------------------------------------------------------------------------
extern "C" void kernel_launch(void* const* d_in, const int* in_sizes, int n_in,
                              void* d_out, int out_size, void* d_ws, size_t ws_size,
                              hipStream_t stream)
{
    (void)in_sizes; (void)n_in; (void)out_size; (void)ws_size;
    const float* x     = (const float*)d_in[0];   // [B][S][D]
    const float* st0   = (const float*)d_in[1];   // [B][N]
    const float* W_B   = (const float*)d_in[2];   // [N][D]
    const float* b_B   = (const float*)d_in[3];   // [N]
    const float* W_C   = (const float*)d_in[4];   // [D][N]
    const float* b_C   = (const float*)d_in[5];   // [D]
    const float* A_log = (const float*)d_in[6];   // [N]

    float* out    = (float*)d_out;                       // [B][S][D]
    float* fstate = out + (size_t)kB * kS * kD;          // [B][N]

    float* buf  = (float*)d_ws;                          // [B][S][N]  (~16.8 MB)
    float* yend = buf  + (size_t)kB * kS * kN;           // [B][NCH][N]
    float* ebuf = yend + (size_t)kB * kNCH * kN;         // [B][NCH][N]

    const int M = kB * kS;   // 16384

    // GEMM1: Ball = x @ W_B^T + b_B    (M x 256, K = 1024)
    gemm_f32_wmma<<<dim3(M / 128, kN / 128), 256, 0, stream>>>(
        x, W_B, b_B, buf, M, kN, kD);

    // Chunked linear scan over S
    scan_local<<<kB * kNCH, kN, 0, stream>>>(A_log, buf, yend);
    scan_carry<<<(kB * kN) / 256, 256, 0, stream>>>(A_log, yend, ebuf);
    scan_apply<<<kB * kNCH, kN, 0, stream>>>(A_log, st0, ebuf, buf, fstate);

    // GEMM2: out = states @ W_C^T + b_C (M x 1024, K = 256)
    gemm_f32_wmma<<<dim3(M / 128, kD / 128), 256, 0, stream>>>(
        buf, W_C, b_C, out, M, kD, kN);
}